// IntraClassConsistencyLoss_22076131901573
// MI455X (gfx1250) — compile-verified
//
#include <hip/hip_runtime.h>

// ---------------------------------------------------------------------------
// IntraClassConsistencyLoss for MI455X (gfx1250, wave32)
//
// loss = BETA * sum_{c: cnt>1} var_c(d) / n_present
//   d_i = || x_i - centroid[label_i] ||^2,  var = unbiased per-class variance
//
// Memory-bound: 2 passes over 128MB of embeddings => ~11us @ 23.3 TB/s.
// Per-class sums via LDS ds_add_f32 tiles + deterministic partial reduction.
// Per-class d stats via f64 global atomics (numerically dominates reference's
// own f32 two-pass variance error).
// Final masked class-sum is done on the matrix unit (v_wmma_f32_16x16x4_f32).
// ---------------------------------------------------------------------------

#define MAXC     128     // capacity for num_classes (runtime nc <= 128)
#define SLICE_W  128     // columns per D-slice in the sums pass
#define CHUNKS   32      // row chunks in the sums pass
#define BETA_F   0.3f

typedef __attribute__((ext_vector_type(2))) float v2f;
typedef __attribute__((ext_vector_type(8))) float v8f;

// ---- K0: zero the atomic accumulators (tiny; partials/sums fully rewritten)
__global__ void icc_zero(double* __restrict__ dsum, double* __restrict__ dsq,
                         int* __restrict__ counts) {
  int t = threadIdx.x;
  if (t < MAXC) { dsum[t] = 0.0; dsq[t] = 0.0; counts[t] = 0; }
}

// ---- K1: class counts (LDS histogram -> few global atomics)
__global__ void icc_count(const int* __restrict__ labels, const int* __restrict__ ncp,
                          int* __restrict__ counts, int N) {
  __shared__ int lc[MAXC];
  int tid = threadIdx.x;
  for (int i = tid; i < MAXC; i += blockDim.x) lc[i] = 0;
  __syncthreads();
  for (int i = blockIdx.x * blockDim.x + tid; i < N; i += gridDim.x * blockDim.x)
    atomicAdd(&lc[labels[i]], 1);
  __syncthreads();
  int nc = *ncp;
  for (int c = tid; c < nc; c += blockDim.x)
    if (lc[c]) atomicAdd(&counts[c], lc[c]);
}

// ---- K2: per-class column sums, one [nc x 128] f32 tile per block in LDS.
// grid = (D/SLICE_W slices, CHUNKS row-chunks), block = 256 (8 waves).
__global__ void icc_partial_sums(const float* __restrict__ E, const int* __restrict__ labels,
                                 const int* __restrict__ ncp, float* __restrict__ partial,
                                 int N, int D) {
  __shared__ float acc[MAXC * SLICE_W];   // 64 KB
  const int nc  = *ncp;
  const int s   = blockIdx.x;             // D slice
  const int ch  = blockIdx.y;             // row chunk
  const int tid = threadIdx.x;
  const int nwords = nc * SLICE_W;
  for (int i = tid; i < nwords; i += blockDim.x) acc[i] = 0.f;
  __syncthreads();

  const int rpc     = N / CHUNKS;
  const int rowBase = ch * rpc;
  const int wave = tid >> 5, lane = tid & 31;
  const int col0 = lane * 4;
  const float4* __restrict__ E4 = reinterpret_cast<const float4*>(E);

  for (int r = rowBase + wave; r < rowBase + rpc; r += 8) {   // 8 waves/block
    float4 x = E4[((size_t)r * D + (size_t)s * SLICE_W + col0) >> 2];
    int c = labels[r];
    float* a = &acc[c * SLICE_W + col0];
    atomicAdd(a + 0, x.x);   // ds_add_f32
    atomicAdd(a + 1, x.y);
    atomicAdd(a + 2, x.z);
    atomicAdd(a + 3, x.w);
  }
  __syncthreads();

  for (int i = tid; i < nwords; i += blockDim.x) {
    int c = i >> 7, col = i & (SLICE_W - 1);
    partial[((size_t)ch * MAXC + c) * D + (size_t)s * SLICE_W + col] = acc[i];
  }
}

// ---- K3: deterministic reduction of partials -> sums[nc][D]
__global__ void icc_reduce_sums(const float* __restrict__ partial, const int* __restrict__ ncp,
                                float* __restrict__ sums, int D) {
  int nc = *ncp;
  size_t idx = (size_t)blockIdx.x * blockDim.x + threadIdx.x;   // = c*D + d
  if (idx >= (size_t)nc * D) return;
  float ssum = 0.f;
  for (int ch = 0; ch < CHUNKS; ++ch)
    ssum += partial[(size_t)ch * MAXC * D + idx];
  sums[idx] = ssum;
}

// ---- K4: d_i = ||x_i - sums[c]*inv_cnt||^2 ; accumulate f64 Σd, Σd² per class.
// One wave per row; block = 256 (8 waves) covers 32 rows.
__global__ void icc_dist(const float* __restrict__ E, const int* __restrict__ labels,
                         const float* __restrict__ sums, const int* __restrict__ counts,
                         double* __restrict__ dsum, double* __restrict__ dsq,
                         int N, int D) {
  const int wave = threadIdx.x >> 5, lane = threadIdx.x & 31;
  const int base = blockIdx.x * 32 + wave * 4;
  const float4* __restrict__ E4 = reinterpret_cast<const float4*>(E);
  const float4* __restrict__ S4 = reinterpret_cast<const float4*>(sums);
  const int nq = D >> 2;   // float4 per row

  for (int rr = 0; rr < 4; ++rr) {
    int r = base + rr;
    if (r >= N) break;
    int c = labels[r];
    float cnt = (float)counts[c];
    float inv = 1.0f / fmaxf(cnt, 1.0f);
    size_t ro = (size_t)r * nq, co = (size_t)c * nq;
    float a = 0.f;
    for (int j = lane; j < nq; j += 32) {
      float4 x = E4[ro + j];
      float4 sv = S4[co + j];           // L2-hot (400 KB working set)
      float dx = x.x - sv.x * inv;
      float dy = x.y - sv.y * inv;
      float dz = x.z - sv.z * inv;
      float dw = x.w - sv.w * inv;
      a += dx * dx + dy * dy + dz * dz + dw * dw;
    }
    for (int off = 16; off > 0; off >>= 1) a += __shfl_xor(a, off, 32);
    if (lane == 0) {
      atomicAdd(&dsum[c], (double)a);                 // global_atomic_add_f64
      atomicAdd(&dsq[c],  (double)a * (double)a);
    }
  }
}

// ---- K5: per-class variance, masked sum over classes on the matrix unit.
// Single wave. Pack w[0..127] into two 16x4 f32 A-matrices, B = ones:
//   D[m][n] = sum_k A[m][k]  =>  total loss = sum of D rows.
__global__ void icc_finalize(const double* __restrict__ dsum, const double* __restrict__ dsq,
                             const int* __restrict__ counts, const int* __restrict__ ncp,
                             float* __restrict__ out) {
  __shared__ float w[MAXC];
  const int lane = threadIdx.x;   // 32 threads, full wave (EXEC all ones at WMMA)
  const int nc = *ncp;
  int npres = 0;
  for (int c = lane; c < MAXC; c += 32) {
    float wv = 0.f;
    if (c < nc) {
      int cnt = counts[c];
      if (cnt > 0) npres++;
      if (cnt > 1) {
        double sd = dsum[c], sq = dsq[c];
        double var = (sq - sd * sd / (double)cnt) / (double)(cnt - 1);
        wv = (float)var;
      }
    }
    w[c] = wv;
  }
  for (int off = 16; off > 0; off >>= 1) npres += __shfl_xor(npres, off, 32);
  __syncthreads();

  // A 16x4 f32 layout: lane L -> M = L%16; VGPR0 holds K = 2*(L/16), VGPR1 K+1.
  const int m = lane & 15, h = lane >> 4;
  v2f a1 = { w[4 * m + 2 * h],      w[4 * m + 2 * h + 1] };
  v2f a2 = { w[64 + 4 * m + 2 * h], w[64 + 4 * m + 2 * h + 1] };
  v2f b  = { 1.0f, 1.0f };          // all-ones B: layout-invariant
  v8f c0 = {};
  c0 = __builtin_amdgcn_wmma_f32_16x16x4_f32(false, a1, false, b, (short)0, c0, false, false);
  c0 = __builtin_amdgcn_wmma_f32_16x16x4_f32(false, a2, false, b, (short)0, c0, false, false);

  // C/D layout: lanes 0-15 hold M=VGPR (N=lane); lanes 16-31 hold M=8+VGPR.
  float S = c0[0] + c0[1] + c0[2] + c0[3] + c0[4] + c0[5] + c0[6] + c0[7];
  float total = __shfl(S, 0, 32) + __shfl(S, 16, 32);
  if (lane == 0)
    out[0] = BETA_F * total / fmaxf((float)npres, 1.0f);
}

// ---------------------------------------------------------------------------
extern "C" void kernel_launch(void* const* d_in, const int* in_sizes, int n_in,
                              void* d_out, int out_size, void* d_ws, size_t ws_size,
                              hipStream_t stream) {
  const float* E      = (const float*)d_in[0];
  const int*   labels = (const int*)d_in[1];
  const int*   ncp    = (const int*)d_in[2];
  float*       out    = (float*)d_out;

  const int N = in_sizes[1];
  const int D = in_sizes[0] / N;   // 1024; assumed multiple of SLICE_W

  char* ws = (char*)d_ws;
  double* dsum   = (double*)(ws + 0);
  double* dsq    = (double*)(ws + 1024);
  int*    counts = (int*)(ws + 2048);
  float*  sums   = (float*)(ws + 4096);
  float*  partial = (float*)(ws + 4096 + (size_t)MAXC * D * sizeof(float));

  icc_zero<<<1, 128, 0, stream>>>(dsum, dsq, counts);
  icc_count<<<32, 256, 0, stream>>>(labels, ncp, counts, N);

  dim3 g2(D / SLICE_W, CHUNKS);
  icc_partial_sums<<<g2, 256, 0, stream>>>(E, labels, ncp, partial, N, D);

  int total = MAXC * D;
  icc_reduce_sums<<<(total + 255) / 256, 256, 0, stream>>>(partial, ncp, sums, D);

  icc_dist<<<(N + 31) / 32, 256, 0, stream>>>(E, labels, sums, counts, dsum, dsq, N, D);

  icc_finalize<<<1, 32, 0, stream>>>(dsum, dsq, counts, ncp, out);
}